// CPABTransformation_2688649527926
// MI455X (gfx1250) — compile-verified
//
#include <hip/hip_runtime.h>
#include <hip/hip_bf16.h>
#include <math.h>

// ---------------------------------------------------------------------------
// CPAB transformation on MI455X (gfx1250, wave32).
//   Kernel 1: single resident workgroup (16 wave32 waves) runs the whole
//             100-step Adam optimization with theta/Trels/grads in LDS (<320KB/WGP), Adam m/v in ws.
//             f32 WMMA (V_WMMA_F32_16X16X4_F32) for theta@B^T and dAvees@B.
//   Kernel 2: throughput pass: 80x16384 points x 50 affine steps; per-batch
//             transforms staged into LDS via GLOBAL_LOAD_ASYNC_TO_LDS_B128
//             (ASYNCcnt / s_wait_asynccnt) when available.
// ---------------------------------------------------------------------------

typedef __attribute__((ext_vector_type(2))) float v2f;
typedef __attribute__((ext_vector_type(8))) float v8f;
typedef __attribute__((ext_vector_type(4))) int   v4i;

#define GLOBAL_AS __attribute__((address_space(1)))
#define LDS_AS    __attribute__((address_space(3)))

typedef GLOBAL_AS v4i* gv4i_ptr;
typedef LDS_AS    v4i* lv4i_ptr;

#define NTHR   512
#define NWAVES 16
#define BS     80
#define P      5
#define D      58
#define DPAD   64
#define DF     384
#define NC     64
#define NSTEP  50
#define MAXIT  100
#define DTSTEP 0.02f
#define LRATE  0.01f

// LDS layout (floats)
#define S_THETA   0
#define S_G       (S_THETA + BS*DPAD)          // 5120
#define S_TRELS   (S_G + BS*DPAD)              // + 5120
#define S_DTRELS  (S_TRELS + BS*NC*6)          // + 30720
#define S_KPS     (S_DTRELS + BS*NC*6)         // + 30720
#define S_KPD     (S_KPS + BS*2*P)             // + 800
#define S_PART    (S_KPD + BS*2*P)             // + 800
#define S_LOSS    (S_PART + NTHR)              // + 512
#define SMEM_FLOATS (S_LOSS + 16)
#define SMEM_BYTES  (SMEM_FLOATS * 4)

// Workspace layout (floats) in d_ws
#define W_AVEES   0                            // 80*384
#define W_M       (W_AVEES + BS*DF)            // 80*58
#define W_V       (W_M + BS*D)
#define W_TRAJ    (W_V + BS*D)                 // 80*5*51*2
#define W_TRELS8  (W_TRAJ + BS*P*(NSTEP+1)*2)  // 80*64*8 padded for b128 loads

__device__ __forceinline__ v8f wmma_f32_16x16x4(v2f a, v2f b, v8f c) {
  return __builtin_amdgcn_wmma_f32_16x16x4_f32(false, a, false, b, (short)0, c,
                                               false, false);
}

__device__ __forceinline__ int findcell(float px, float py) {
  float xc = fminf(fmaxf(px * 4.0f, 0.0f), 4.0f - 1e-6f);
  float yc = fminf(fmaxf(py * 4.0f, 0.0f), 4.0f - 1e-6f);
  float ixf = floorf(xc), iyf = floorf(yc);
  float xr = xc - ixf, yr = yc - iyf;
  int tri = (xr < yr) ? ((1.0f - xr < yr) ? 2 : 3)
                      : ((1.0f - xr < yr) ? 1 : 0);
  return ((int)iyf * 4 + (int)ixf) * 4 + tri;
}

// expm of [[M],[0,0,0]] (3x3) via 12-term Taylor; M,R stored as 2x3 row-major.
__device__ __forceinline__ void expm23(const float* M, float* R) {
  float t0 = 1.f, t1 = 0.f, t2 = 0.f, t3 = 0.f, t4 = 1.f, t5 = 0.f;
  float r0 = 1.f, r1 = 0.f, r2 = 0.f, r3 = 0.f, r4 = 1.f, r5 = 0.f;
#pragma unroll
  for (int k = 1; k <= 12; ++k) {
    float inv = 1.0f / (float)k;
    float n0 = (t0 * M[0] + t1 * M[3]) * inv;
    float n1 = (t0 * M[1] + t1 * M[4]) * inv;
    float n2 = (t0 * M[2] + t1 * M[5]) * inv;
    float n3 = (t3 * M[0] + t4 * M[3]) * inv;
    float n4 = (t3 * M[1] + t4 * M[4]) * inv;
    float n5 = (t3 * M[2] + t4 * M[5]) * inv;
    t0 = n0; t1 = n1; t2 = n2; t3 = n3; t4 = n4; t5 = n5;
    r0 += t0; r1 += t1; r2 += t2; r3 += t3; r4 += t4; r5 += t5;
  }
  R[0] = r0; R[1] = r1; R[2] = r2; R[3] = r3; R[4] = r4; R[5] = r5;
}

// Adjoint of expm23: given dR (2x3) and M (2x3), produce dM (2x3).
__device__ __forceinline__ void expm23_bwd(const float* M, const float* dR,
                                           float* dM) {
  float Ts[13][6];
  Ts[0][0] = 1.f; Ts[0][1] = 0.f; Ts[0][2] = 0.f;
  Ts[0][3] = 0.f; Ts[0][4] = 1.f; Ts[0][5] = 0.f;
#pragma unroll
  for (int k = 1; k <= 12; ++k) {
    float inv = 1.0f / (float)k;
    const float* Pm = Ts[k - 1];
    float* Q = Ts[k];
    Q[0] = (Pm[0] * M[0] + Pm[1] * M[3]) * inv;
    Q[1] = (Pm[0] * M[1] + Pm[1] * M[4]) * inv;
    Q[2] = (Pm[0] * M[2] + Pm[1] * M[5]) * inv;
    Q[3] = (Pm[3] * M[0] + Pm[4] * M[3]) * inv;
    Q[4] = (Pm[3] * M[1] + Pm[4] * M[4]) * inv;
    Q[5] = (Pm[3] * M[2] + Pm[4] * M[5]) * inv;
  }
  float c0 = 0.f, c1 = 0.f, c3 = 0.f, c4 = 0.f;
  dM[0] = dM[1] = dM[2] = dM[3] = dM[4] = dM[5] = 0.f;
#pragma unroll
  for (int k = 12; k >= 1; --k) {
    float g0 = dR[0] + c0, g1 = dR[1] + c1, g2 = dR[2];
    float g3 = dR[3] + c3, g4 = dR[4] + c4, g5 = dR[5];
    float inv = 1.0f / (float)k;
    const float* Pm = Ts[k - 1];
    dM[0] += (Pm[0] * g0 + Pm[3] * g3) * inv;
    dM[1] += (Pm[0] * g1 + Pm[3] * g4) * inv;
    dM[2] += (Pm[0] * g2 + Pm[3] * g5) * inv;
    dM[3] += (Pm[1] * g0 + Pm[4] * g3) * inv;
    dM[4] += (Pm[1] * g1 + Pm[4] * g4) * inv;
    dM[5] += (Pm[1] * g2 + Pm[4] * g5) * inv;
    c0 = (g0 * M[0] + g1 * M[1] + g2 * M[2]) * inv;
    c1 = (g0 * M[3] + g1 * M[4] + g2 * M[5]) * inv;
    c3 = (g3 * M[0] + g4 * M[1] + g5 * M[2]) * inv;
    c4 = (g3 * M[3] + g4 * M[4] + g5 * M[5]) * inv;
  }
}

__global__ void __launch_bounds__(NTHR)
cpab_adam_kernel(const float* __restrict__ kps_in,
                 const float* __restrict__ kpd_in,
                 const float* __restrict__ Bm,
                 float* __restrict__ ws) {
  extern __shared__ float smem[];
  float* sTheta  = smem + S_THETA;
  float* sG      = smem + S_G;
  float* sTrels  = smem + S_TRELS;
  float* sdTrels = smem + S_DTRELS;
  float* sKps    = smem + S_KPS;
  float* sKpd    = smem + S_KPD;
  float* sPart   = smem + S_PART;
  float* sLoss   = smem + S_LOSS;

  float* wsAvees = ws + W_AVEES;
  float* wsM     = ws + W_M;
  float* wsV     = ws + W_V;
  float* wsTraj  = ws + W_TRAJ;
  float* wsTr8   = ws + W_TRELS8;

  const int tid  = threadIdx.x;
  const int wave = tid >> 5;
  const int lane = tid & 31;
  const int l16  = lane & 15;
  const int half = lane >> 4;

  // ---- one-time init (every launch; ws is not trusted across calls) ----
  for (int e = tid; e < BS * DPAD; e += NTHR) {
    int col = e & (DPAD - 1);
    sTheta[e] = (col < D) ? 1e-6f : 0.0f;
  }
  for (int e = tid; e < BS * D; e += NTHR) { wsM[e] = 0.f; wsV[e] = 0.f; }
  for (int e = tid; e < BS * 2 * P; e += NTHR) {
    int bb = e / (2 * P), r = e % (2 * P);
    int c = r / P, p = r % P;
    float vs = kps_in[(bb * P + p) * 2 + c];
    float vd = kpd_in[(bb * P + p) * 2 + c];
    sKps[e] = vs;
    sKpd[e] = (vd + 1.0f) * 0.5f;
  }
  __threadfence();
  __syncthreads();

  for (int it = 1; it <= MAXIT; ++it) {
    // ---- phase a: zero accumulators ----
    for (int e = tid; e < BS * NC * 6; e += NTHR) sdTrels[e] = 0.f;
    sPart[tid] = 0.f;
    if (tid == 0) sLoss[0] = 0.f;
    __syncthreads();

    // ---- phase b: GEMM1  Avees[80,384] = theta[80,58] @ B^T  (WMMA f32) ----
    for (int tile = wave; tile < 5 * 24; tile += NWAVES) {
      int mt = tile / 24, nt = tile % 24;
      int m = mt * 16 + l16;
      int n = nt * 16 + l16;
      v8f c = {};
      for (int ks = 0; ks < 15; ++ks) {
        int k0 = ks * 4 + 2 * half;
        v2f a, b;
        a.x = sTheta[m * DPAD + k0];
        a.y = sTheta[m * DPAD + k0 + 1];
        b.x = (k0 < D) ? Bm[n * D + k0] : 0.f;
        b.y = (k0 + 1 < D) ? Bm[n * D + k0 + 1] : 0.f;
        c = wmma_f32_16x16x4(a, b, c);
      }
      union { v8f v; float f[8]; } cu; cu.v = c;
#pragma unroll
      for (int r = 0; r < 8; ++r) {
        int mm = mt * 16 + r + 8 * half;
        wsAvees[mm * DF + n] = cu.f[r];
      }
    }
    __threadfence();
    __syncthreads();

    // ---- phase c: expm forward -> sTrels ----
    for (int cell = tid; cell < BS * NC; cell += NTHR) {
      int bb = cell >> 6, cc = cell & 63;
      float M[6], R[6];
#pragma unroll
      for (int j = 0; j < 6; ++j) M[j] = DTSTEP * wsAvees[bb * DF + cc * 6 + j];
      expm23(M, R);
#pragma unroll
      for (int j = 0; j < 6; ++j) sTrels[cell * 6 + j] = R[j];
    }
    __syncthreads();

    // ---- phase d: forward scan over 400 keypoints, residual partials ----
    if (tid < BS * P) {
      int bb = tid / P, p = tid % P;
      float x = sKpd[bb * 10 + p];
      float y = sKpd[bb * 10 + P + p];
      float* tr = wsTraj + (size_t)tid * ((NSTEP + 1) * 2);
      tr[0] = x; tr[1] = y;
#pragma unroll 1
      for (int s = 0; s < NSTEP; ++s) {
        int idx = findcell(x, y);
        const float* T = &sTrels[(bb * NC + idx) * 6];
        float nx = T[0] * x + T[1] * y + T[2];
        float ny = T[3] * x + T[4] * y + T[5];
        x = nx; y = ny;
        tr[(s + 1) * 2] = x; tr[(s + 1) * 2 + 1] = y;
      }
      float rx = (2.f * x - 1.f) - sKps[bb * 10 + p];
      float ry = (2.f * y - 1.f) - sKps[bb * 10 + P + p];
      sPart[tid] = rx * rx + ry * ry;
    }
    __syncthreads();
    for (int s = NTHR / 2; s > 0; s >>= 1) {
      if (tid < s) sPart[tid] += sPart[tid + s];
      __syncthreads();
    }
    if (tid == 0) sLoss[0] = sqrtf(sPart[0]);
    __syncthreads();

    // ---- phase e: backward scan, scatter-accumulate dTrels (LDS atomics) ----
    if (tid < BS * P) {
      int bb = tid / P, p = tid % P;
      const float* tr = wsTraj + (size_t)tid * ((NSTEP + 1) * 2);
      float loss = sLoss[0];
      float xf = tr[NSTEP * 2], yf = tr[NSTEP * 2 + 1];
      float dx = 2.f * ((2.f * xf - 1.f) - sKps[bb * 10 + p]) / loss;
      float dy = 2.f * ((2.f * yf - 1.f) - sKps[bb * 10 + P + p]) / loss;
#pragma unroll 1
      for (int s = NSTEP - 1; s >= 0; --s) {
        float x = tr[s * 2], y = tr[s * 2 + 1];
        int idx = findcell(x, y);
        float* dT = &sdTrels[(bb * NC + idx) * 6];
        atomicAdd(&dT[0], dx * x); atomicAdd(&dT[1], dx * y); atomicAdd(&dT[2], dx);
        atomicAdd(&dT[3], dy * x); atomicAdd(&dT[4], dy * y); atomicAdd(&dT[5], dy);
        const float* T = &sTrels[(bb * NC + idx) * 6];
        float ndx = T[0] * dx + T[3] * dy;
        float ndy = T[1] * dx + T[4] * dy;
        dx = ndx; dy = ndy;
      }
    }
    __syncthreads();

    // ---- phase f: expm backward, dAvees overwrites wsAvees ----
    for (int cell = tid; cell < BS * NC; cell += NTHR) {
      int bb = cell >> 6, cc = cell & 63;
      float M[6], dR[6], dM[6];
#pragma unroll
      for (int j = 0; j < 6; ++j) M[j] = DTSTEP * wsAvees[bb * DF + cc * 6 + j];
#pragma unroll
      for (int j = 0; j < 6; ++j) dR[j] = sdTrels[cell * 6 + j];
      expm23_bwd(M, dR, dM);
#pragma unroll
      for (int j = 0; j < 6; ++j) wsAvees[bb * DF + cc * 6 + j] = DTSTEP * dM[j];
    }
    __threadfence();
    __syncthreads();

    // ---- phase g: GEMM2  g[80,58] = dAvees[80,384] @ B  (WMMA f32) ----
    for (int tile = wave; tile < 5 * 4; tile += NWAVES) {
      int mt = tile / 4, nt = tile % 4;
      int m = mt * 16 + l16;
      int n = nt * 16 + l16;
      v8f c = {};
      for (int ks = 0; ks < 96; ++ks) {
        int k0 = ks * 4 + 2 * half;
        v2f a, b;
        a.x = wsAvees[m * DF + k0];
        a.y = wsAvees[m * DF + k0 + 1];
        b.x = (n < D) ? Bm[k0 * D + n] : 0.f;
        b.y = (n < D) ? Bm[(k0 + 1) * D + n] : 0.f;
        c = wmma_f32_16x16x4(a, b, c);
      }
      union { v8f v; float f[8]; } cu; cu.v = c;
#pragma unroll
      for (int r = 0; r < 8; ++r) {
        int mm = mt * 16 + r + 8 * half;
        sG[mm * DPAD + n] = cu.f[r];
      }
    }
    __syncthreads();

    // ---- phase h: Adam update (m,v owned per-thread across iterations) ----
    float b1t = powf(0.9f, (float)it);
    float b2t = powf(0.999f, (float)it);
    for (int e = tid; e < BS * D; e += NTHR) {
      int bb = e / D, j = e % D;
      float g = sG[bb * DPAD + j];
      float m = 0.9f * wsM[e] + 0.1f * g;
      float v = 0.999f * wsV[e] + 0.001f * g * g;
      wsM[e] = m; wsV[e] = v;
      float mh = m / (1.0f - b1t);
      float vh = v / (1.0f - b2t);
      sTheta[bb * DPAD + j] -= LRATE * mh / (sqrtf(vh) + 1e-8f);
    }
    __syncthreads();
  }

  // ---- final: recompute Avees/Trels with optimized theta; pad to 8 ----
  for (int tile = wave; tile < 5 * 24; tile += NWAVES) {
    int mt = tile / 24, nt = tile % 24;
    int m = mt * 16 + l16;
    int n = nt * 16 + l16;
    v8f c = {};
    for (int ks = 0; ks < 15; ++ks) {
      int k0 = ks * 4 + 2 * half;
      v2f a, b;
      a.x = sTheta[m * DPAD + k0];
      a.y = sTheta[m * DPAD + k0 + 1];
      b.x = (k0 < D) ? Bm[n * D + k0] : 0.f;
      b.y = (k0 + 1 < D) ? Bm[n * D + k0 + 1] : 0.f;
      c = wmma_f32_16x16x4(a, b, c);
    }
    union { v8f v; float f[8]; } cu; cu.v = c;
#pragma unroll
    for (int r = 0; r < 8; ++r) {
      int mm = mt * 16 + r + 8 * half;
      wsAvees[mm * DF + n] = cu.f[r];
    }
  }
  __threadfence();
  __syncthreads();
  for (int cell = tid; cell < BS * NC; cell += NTHR) {
    int bb = cell >> 6, cc = cell & 63;
    float M[6], R[6];
#pragma unroll
    for (int j = 0; j < 6; ++j) M[j] = DTSTEP * wsAvees[bb * DF + cc * 6 + j];
    expm23(M, R);
#pragma unroll
    for (int j = 0; j < 6; ++j) wsTr8[cell * 8 + j] = R[j];
    wsTr8[cell * 8 + 6] = 0.f; wsTr8[cell * 8 + 7] = 0.f;
  }
}

// ---- Kernel 2: final 128x128 grid transform, one thread per point --------
__global__ void __launch_bounds__(256)
cpab_grid_kernel(const float* __restrict__ trels8, float* __restrict__ out) {
  __shared__ float sT[NC * 8];
  const int b = blockIdx.y;

#if defined(__gfx1250__) && \
    __has_builtin(__builtin_amdgcn_global_load_async_to_lds_b128)
  // CDNA5 async DMA: global -> LDS without VGPR round trip (ASYNCcnt).
  if (threadIdx.x < (NC * 8) / 4) {
    const float* gp = trels8 + (size_t)b * NC * 8 + threadIdx.x * 4;
    float* lp = &sT[threadIdx.x * 4];
    __builtin_amdgcn_global_load_async_to_lds_b128(
        (gv4i_ptr)(GLOBAL_AS void*)gp, (lv4i_ptr)(LDS_AS void*)lp,
        /*offset=*/0, /*cpol=*/0);
  }
#if __has_builtin(__builtin_amdgcn_s_wait_asynccnt)
  __builtin_amdgcn_s_wait_asynccnt(0);
#else
  asm volatile("s_wait_asynccnt 0x0" ::: "memory");
#endif
#else
  for (int i = threadIdx.x; i < NC * 8; i += blockDim.x)
    sT[i] = trels8[(size_t)b * NC * 8 + i];
#endif
  __syncthreads();

  int pix = blockIdx.x * blockDim.x + threadIdx.x;  // 0..16383
  float x = (float)(pix & 127) * (1.0f / 127.0f);
  float y = (float)(pix >> 7) * (1.0f / 127.0f);
#pragma unroll 1
  for (int s = 0; s < NSTEP; ++s) {
    int idx = findcell(x, y);
    const float4 t0 = *(const float4*)&sT[idx * 8];      // ds_load_b128
    const float2 t1 = *(const float2*)&sT[idx * 8 + 4];  // ds_load_b64
    float nx = t0.x * x + t0.y * y + t0.z;
    float ny = t0.w * x + t1.x * y + t1.y;
    x = nx; y = ny;
  }
  size_t o = ((size_t)b * 16384 + (size_t)pix) * 2;
  out[o]     = 2.0f * x - 1.0f;
  out[o + 1] = 2.0f * y - 1.0f;
}

extern "C" void kernel_launch(void* const* d_in, const int* in_sizes, int n_in,
                              void* d_out, int out_size, void* d_ws,
                              size_t ws_size, hipStream_t stream) {
  // inputs: [0]=source_image(f32, unused beyond shape), [1]=kp_source,
  //         [2]=kp_driving, [3]=num_of_pts(int,==5), [4]=B [384,58] f32
  const float* kps = (const float*)d_in[1];
  const float* kpd = (const float*)d_in[2];
  const float* Bm  = (const float*)d_in[4];
  float* ws  = (float*)d_ws;
  float* out = (float*)d_out;

  cpab_adam_kernel<<<1, NTHR, SMEM_BYTES, stream>>>(kps, kpd, Bm, ws);

  dim3 g2(64, BS);  // 64 blocks of 256 pixels per batch, 80 batches
  cpab_grid_kernel<<<g2, 256, 0, stream>>>(ws + W_TRELS8, out);
}